// SNN_test_78735340471025
// MI455X (gfx1250) — compile-verified
//
#include <hip/hip_runtime.h>

// LIF scan for MI455X (gfx1250, wave32).
// Pass 1: parallel exact linear scan of i_t using V_WMMA_F32_16X16X4_F32
//         (16x16 lower-triangular decay matrix x 16 chunks of 16 inputs).
// Pass 2: f32-exact cross-segment carry fixup (a^4096 underflows -> segments decouple).
// Pass 3: exact sequential v/s recurrence, one wave, shuffle-broadcast + coalesced I/O.

typedef __attribute__((ext_vector_type(2))) float v2f;
typedef __attribute__((ext_vector_type(8))) float v8f;

#define LEAK_I_C (6.0f / 7.0f)   // 1 - 1/tau_syn
#define LEAK_V_C 0.9f            // 1 - 1/tau_mem
#define LOG2_LEAK_I (-0.22239242133644802f)  // log2(6/7)
#define SEGLEN   4096            // timesteps per wave (a^4096 == 0 in f32)
#define TILE     256             // timesteps per WMMA tile (16 chunks x 16)
#define TILES_PER_SEG (SEGLEN / TILE)

// ---------------------------------------------------------------------------
// K1: per-wave segment scan of i_t = a*i_{t-1} + w*x_t  (zero-init per segment)
//     i written to d_out states slot [T + 3*t + 2]; segment end value -> ws.
// ---------------------------------------------------------------------------
__global__ __launch_bounds__(256) void lif_i_scan(const float* __restrict__ x,
                                                  const float* __restrict__ weight,
                                                  float* __restrict__ out,
                                                  float* __restrict__ segend,
                                                  int T) {
  const int lane = threadIdx.x & 31;
  const int wid  = (int)((blockIdx.x * (unsigned)blockDim.x + threadIdx.x) >> 5);
  const int n    = lane & 15;   // chunk index (B/C/D column), also A row
  const int hi   = lane >> 4;   // lane-half (K/M split per ISA VGPR layout)
  const float w  = weight[0];

  // Exact decay power table a^0..a^16.
  float ap[17];
  ap[0] = 1.0f;
#pragma unroll
  for (int p = 1; p <= 16; ++p) ap[p] = ap[p - 1] * LEAK_I_C;
  const float a16 = ap[16];
  const float a32 = a16 * a16, a64 = a32 * a32, a128 = a64 * a64, a256 = a128 * a128;

  float a16n = 1.0f;                       // a^(16*n), per-lane
  for (int p = 0; p < n; ++p) a16n *= a16;

  // A operand: L[m][j] = (j<=m) ? a^(m-j) : 0, m = n.
  // 16x4 f32 layout: VGPR0 holds K=0 (lanes 0-15) / K=2 (lanes 16-31), VGPR1 K=1/K=3.
  v2f Aq[4];
#pragma unroll
  for (int q = 0; q < 4; ++q) {
    const int j0 = 4 * q + 2 * hi;
    Aq[q].x = (j0     <= n) ? ap[n - j0]     : 0.0f;
    Aq[q].y = (j0 + 1 <= n) ? ap[n - j0 - 1] : 0.0f;
  }
  // a^(k+1) for this lane's C/D rows k = r + 8*hi.
  float pk[8];
#pragma unroll
  for (int r = 0; r < 8; ++r) pk[r] = ap[r + 8 * hi + 1];

  const long segbase = (long)wid * SEGLEN;
  float seg_carry = 0.0f;                       // i value at end of previous tile
  float* __restrict__ iout = out + (long)T;     // states base

  for (int tile = 0; tile < TILES_PER_SEG; ++tile) {
    const long tb = segbase + (long)tile * TILE;
    if (tile + 1 < TILES_PER_SEG)
      __builtin_prefetch(x + tb + TILE + n * 16, 0, 0);   // global_prefetch_b8

    // D = L x U : four chained f32 WMMAs (K = 4 each).
    // B 4x16 layout: VGPR0 = K row (0 | 2hi), VGPR1 = next row -> one float2 load.
    v8f c = {};
#pragma unroll
    for (int q = 0; q < 4; ++q) {
      const float* p = x + tb + (long)n * 16 + 4 * q + 2 * hi;
      v2f u;
      u.x = w * p[0];
      u.y = w * p[1];
      c = __builtin_amdgcn_wmma_f32_16x16x4_f32(false, Aq[q], false, u,
                                                (short)0, c, false, false);
    }

    // Chunk end values e_n = D[15][n] live in lanes 16..31, VGPR 7.
    float e = __shfl(c[7], 16 + n, 32);
    // Weighted inclusive scan over 16 chunks (ratio a^16), both halves identical.
    float incl = e, o;
    o = __shfl_up(incl, 1, 16); if (n >= 1) incl = fmaf(a16,  o, incl);
    o = __shfl_up(incl, 2, 16); if (n >= 2) incl = fmaf(a32,  o, incl);
    o = __shfl_up(incl, 4, 16); if (n >= 4) incl = fmaf(a64,  o, incl);
    o = __shfl_up(incl, 8, 16); if (n >= 8) incl = fmaf(a128, o, incl);
    float excl = __shfl_up(incl, 1, 16);
    if (n == 0) excl = 0.0f;
    const float tile_end = __shfl(incl, 15, 16);

    // i value just before this chunk: within-tile carry + decayed tile carry-in.
    const float c_eff = fmaf(a16n, seg_carry, excl);
#pragma unroll
    for (int r = 0; r < 8; ++r) {
      const int  k = r + 8 * hi;
      const long g = tb + (long)n * 16 + k;
      iout[3 * g + 2] = fmaf(pk[r], c_eff, c[r]);
    }
    seg_carry = fmaf(a256, seg_carry, tile_end);
  }
  if (lane == 0) segend[wid] = seg_carry;
}

// ---------------------------------------------------------------------------
// K2: cross-segment fixup. a^4096 == 0 in f32, so carry into segment s is
//     exactly segend[s-1] (segment 0: initial state i). a^(t+1) == 0.0f for
//     t >= ~700, so touching the first 1024 elements is f32-exact.
// ---------------------------------------------------------------------------
__global__ __launch_bounds__(256) void lif_i_fixup(const float* __restrict__ state,
                                                   const float* __restrict__ segend,
                                                   float* __restrict__ out,
                                                   int T) {
  const int seg = blockIdx.x;
  const float carry = (seg == 0) ? state[2] : segend[seg - 1];
  if (carry == 0.0f) return;
  float* __restrict__ iout = out + (long)T;
  const long base = (long)seg * SEGLEN;
  for (int t = threadIdx.x; t < 1024; t += blockDim.x) {
    const float f = __builtin_exp2f((float)(t + 1) * LOG2_LEAK_I);   // a^(t+1)
    const long g = base + t;
    iout[3 * g + 2] = fmaf(f, carry, iout[3 * g + 2]);
  }
}

// ---------------------------------------------------------------------------
// K3: exact sequential v/s recurrence. Single wave; 32 coalesced i-loads per
//     outer iteration (L2-resident), 32 serial steps with readlane broadcast,
//     coalesced s/v stores. Chain per step: fma -> cmp -> cndmask.
// ---------------------------------------------------------------------------
__global__ __launch_bounds__(32) void lif_vs_seq(const float* __restrict__ state,
                                                 float* __restrict__ out,
                                                 int T) {
  const int lane = threadIdx.x & 31;
  const float* __restrict__ iarr = out + (long)T;

  float v = state[1];
  bool spike = state[0] > 0.5f;

  for (long t0 = 0; t0 < (long)T; t0 += 32) {
    const long g0 = t0 + lane;
    const float iv = iarr[3 * g0 + 2];
    float myv = 0.0f, mys = 0.0f;
#pragma unroll
    for (int k = 0; k < 32; ++k) {
      const float ik = __shfl(iv, k, 32);
      v = spike ? ik : fmaf(LEAK_V_C, v, ik);   // reset-by-spike
      spike = (v - 1.0f) > 0.0f;                // Heaviside threshold
      if (k == lane) { myv = v; mys = spike ? 1.0f : 0.0f; }
    }
    out[g0] = mys;                  // outputs (T,1)
    out[(long)T + 3 * g0 + 0] = mys;  // states[...,0] = s
    out[(long)T + 3 * g0 + 1] = myv;  // states[...,1] = v
  }
}

// ---------------------------------------------------------------------------
extern "C" void kernel_launch(void* const* d_in, const int* in_sizes, int n_in,
                              void* d_out, int out_size, void* d_ws, size_t ws_size,
                              hipStream_t stream) {
  const float* x      = (const float*)d_in[0];
  const float* state  = (const float*)d_in[1];
  const float* weight = (const float*)d_in[2];
  float* out          = (float*)d_out;
  float* segend       = (float*)d_ws;          // nseg floats (8 KB)

  const int T    = in_sizes[0];                // 8388608
  const int nseg = T / SEGLEN;                 // 2048 waves
  const int nblk = nseg / 8;                   // 8 waves per 256-thread block

  lif_i_scan<<<nblk, 256, 0, stream>>>(x, weight, out, segend, T);
  lif_i_fixup<<<nseg, 256, 0, stream>>>(state, segend, out, T);
  lif_vs_seq<<<1, 32, 0, stream>>>(state, out, T);
}